// All_Graph_Net_9457517986561
// MI455X (gfx1250) — compile-verified
//
#include <hip/hip_runtime.h>

typedef __attribute__((ext_vector_type(2))) float v2f;
typedef __attribute__((ext_vector_type(8))) float v8f;

#define HID 128

__global__ void zero_kernel(float* __restrict__ p, int n) {
    int i = blockIdx.x * blockDim.x + threadIdx.x;
    if (i < n) p[i] = 0.0f;
}

// One 256-thread block (8 wave32) computes a 16x128 output tile.
// out[row, 0:128] = relu(A @ W1 (+ A2 @ W2) + bias), either written or accumulated (+=).
// K = 128 per GEMM, done as 32 chained V_WMMA_F32_16X16X4_F32 per 16x16 tile.
__global__ void __launch_bounds__(256)
sage_gemm_kernel(const float* __restrict__ A, const float* __restrict__ W1,
                 const float* __restrict__ A2, const float* __restrict__ W2,
                 const float* __restrict__ bias, float* __restrict__ out,
                 int n_rows, int accumulate)
{
    const int lane = threadIdx.x & 31;
    const int wave = threadIdx.x >> 5;
    const int row0 = blockIdx.x << 4;
    const int col0 = wave << 4;
    const int half = lane >> 4;   // lanes 0-15: K%4 in {0,1}; lanes 16-31: K%4 in {2,3}
    const int l16  = lane & 15;

    int arow = row0 + l16;
    if (arow >= n_rows) arow = n_rows - 1;   // clamp loads; stores masked in epilogue

    v8f c = {};

    const float* aptr = A + arow * HID;
    #pragma unroll
    for (int kb = 0; kb < HID; kb += 4) {
        const int k = kb + half * 2;
        v2f a, b;
        a.x = aptr[k];
        a.y = aptr[k + 1];
        b.x = W1[k * HID + col0 + l16];
        b.y = W1[(k + 1) * HID + col0 + l16];
        c = __builtin_amdgcn_wmma_f32_16x16x4_f32(false, a, false, b, (short)0, c,
                                                  false, false);
    }
    if (A2 != nullptr) {   // uniform branch: EXEC stays all-ones for WMMA
        const float* a2ptr = A2 + arow * HID;
        #pragma unroll
        for (int kb = 0; kb < HID; kb += 4) {
            const int k = kb + half * 2;
            v2f a, b;
            a.x = a2ptr[k];
            a.y = a2ptr[k + 1];
            b.x = W2[k * HID + col0 + l16];
            b.y = W2[(k + 1) * HID + col0 + l16];
            c = __builtin_amdgcn_wmma_f32_16x16x4_f32(false, a, false, b, (short)0, c,
                                                      false, false);
        }
    }

    // C/D layout: VGPR r, lanes 0-15 -> M=r; lanes 16-31 -> M=8+r; N = lane%16
    const float bv = bias[col0 + l16];
    #pragma unroll
    for (int r = 0; r < 8; ++r) {
        const int row = row0 + r + half * 8;
        if (row < n_rows) {
            float v = c[r] + bv;
            v = v > 0.0f ? v : 0.0f;
            float* o = out + row * HID + col0 + l16;
            if (accumulate) *o += v;
            else            *o  = v;
        }
    }
}

// segment_max over non-negative (post-relu) values: uint bit-pattern atomicMax
// against a zero-initialized buffer == max with DGL "isolated -> 0" semantics.
// 32 threads per edge, float4 per thread (128 features).
__global__ void __launch_bounds__(256)
scatter_max_kernel(const float* __restrict__ pooled, const int* __restrict__ src,
                   const int* __restrict__ dst, unsigned int* __restrict__ agg,
                   int n_edge)
{
    const int tid = blockIdx.x * blockDim.x + threadIdx.x;
    const int e = tid >> 5;
    if (e >= n_edge) return;
    const int f = (tid & 31) << 2;
    const int s = src[e];
    const int d = dst[e];
    const float4 v = *(const float4*)(pooled + s * HID + f);
    unsigned int* ap = agg + d * HID + f;
    atomicMax(ap + 0, __float_as_uint(v.x));
    atomicMax(ap + 1, __float_as_uint(v.y));
    atomicMax(ap + 2, __float_as_uint(v.z));
    atomicMax(ap + 3, __float_as_uint(v.w));
}

extern "C" void kernel_launch(void* const* d_in, const int* in_sizes, int n_in,
                              void* d_out, int out_size, void* d_ws, size_t ws_size,
                              hipStream_t stream) {
    (void)n_in; (void)out_size; (void)ws_size;

    // node types: 0=drug, 1=protein, 2=disease
    static const int rel_st[8] = {2, 2, 2, 0, 0, 1, 0, 1};
    static const int rel_dt[8] = {0, 0, 1, 2, 2, 2, 0, 1};

    const int nn[3] = {in_sizes[0] / HID, in_sizes[1] / HID, in_sizes[2] / HID};

    // d_out: [h1_drug | h1_prot | h2_drug | h2_prot]
    float* h1_drug = (float*)d_out;
    float* h1_prot = h1_drug + (size_t)nn[0] * HID;
    float* h2_drug = h1_prot + (size_t)nn[1] * HID;
    float* h2_prot = h2_drug + (size_t)nn[0] * HID;

    // workspace: pooled (max n_src rows), agg (max n_dst rows), h1_disease
    const int nmax = nn[1] > nn[0] ? (nn[1] > nn[2] ? nn[1] : nn[2])
                                   : (nn[0] > nn[2] ? nn[0] : nn[2]);
    float* pooled = (float*)d_ws;
    float* aggf   = pooled + (size_t)nmax * HID;
    float* h1_dis = aggf   + (size_t)nmax * HID;

    auto zero = [&](float* p, int n) {
        zero_kernel<<<dim3((n + 255) / 256), dim3(256), 0, stream>>>(p, n);
    };

    auto run_rel = [&](int r, const float* const* hin, float* const* hout) {
        const int st = rel_st[r], dt = rel_dt[r];
        const int nsrc = nn[st], ndst = nn[dt];
        const float* Wp = (const float*)d_in[3 + 5 * r + 0];
        const float* bp = (const float*)d_in[3 + 5 * r + 1];
        const float* Ws = (const float*)d_in[3 + 5 * r + 2];
        const float* Wn = (const float*)d_in[3 + 5 * r + 3];
        const float* b  = (const float*)d_in[3 + 5 * r + 4];
        const int* srcI = (const int*)d_in[43 + 2 * r];
        const int* dstI = (const int*)d_in[44 + 2 * r];
        const int nE = in_sizes[43 + 2 * r];

        // pooled = relu(h_src @ Wp + bp)
        sage_gemm_kernel<<<dim3((nsrc + 15) / 16), dim3(256), 0, stream>>>(
            hin[st], Wp, nullptr, nullptr, bp, pooled, nsrc, 0);
        // agg = segment_max(pooled[src], dst) with isolated -> 0
        zero(aggf, ndst * HID);
        const int nthr = nE * 32;
        scatter_max_kernel<<<dim3((nthr + 255) / 256), dim3(256), 0, stream>>>(
            pooled, srcI, dstI, (unsigned int*)aggf, nE);
        // out[dt] += relu(h_dst @ Ws + agg @ Wn + b)
        sage_gemm_kernel<<<dim3((ndst + 15) / 16), dim3(256), 0, stream>>>(
            hin[dt], Ws, aggf, Wn, b, hout[dt], ndst, 1);
    };

    // ---- layer 1: all 8 relations ----
    {
        const float* hin[3]  = {(const float*)d_in[0], (const float*)d_in[1],
                                (const float*)d_in[2]};
        float* hout[3] = {h1_drug, h1_prot, h1_dis};
        zero(h1_drug, nn[0] * HID);
        zero(h1_prot, nn[1] * HID);
        zero(h1_dis,  nn[2] * HID);
        for (int r = 0; r < 8; ++r) run_rel(r, hin, hout);
    }

    // ---- layer 2: only relations whose dst is drug/protein (disease h2 unused) ----
    {
        const float* hin[3]  = {h1_drug, h1_prot, h1_dis};
        float* hout[3] = {h2_drug, h2_prot, nullptr};
        zero(h2_drug, nn[0] * HID);
        zero(h2_prot, nn[1] * HID);
        static const int l2_rels[5] = {0, 1, 2, 6, 7};
        for (int i = 0; i < 5; ++i) run_rel(l2_rels[i], hin, hout);
    }
}